// Sampler_67654324846949
// MI455X (gfx1250) — compile-verified
//
#include <hip/hip_runtime.h>
#include <hip/hip_bf16.h>
#include <math.h>

// ---------------------------------------------------------------------------
// Sampler for MI455X (gfx1250, wave32).
//  - one workgroup (1024 threads = 32 waves) per batch row
//  - logits streamed global -> LDS with gfx1250 async copies
//    (global_load_async_to_lds_b128 + s_wait_asynccnt), 4-deep pipeline
//  - penalized logits cached in d_ws (if big enough) so passes 2-5 skip the
//    per-element binary search over the sorted prev-token list
//  - typical-p / top-k thresholds via two-level LDS histograms (no argsort)
//  - exact JAX threefry2x32 gumbel-max for the final categorical draw
// ---------------------------------------------------------------------------

#define DEVINL __device__ __forceinline__

constexpr int   B_      = 128;
constexpr int   V_      = 128000;
constexpr int   S_      = 2048;
constexpr float TEMP_   = 0.8f;
constexpr float INVT_   = 1.0f / TEMP_;
constexpr int   TOPK_   = 50;
constexpr float TOPP_   = 0.9f;
constexpr float MINP_   = 0.05f;
constexpr float TYPP_   = 0.95f;
constexpr float REP_    = 1.1f;
constexpr float FREQ_   = 0.2f;
constexpr float PRES_   = 0.1f;

constexpr int   NTHREADS = 1024;          // 32 waves on wave32
constexpr int   NWAVES   = NTHREADS / 32;
constexpr int   DEPTH_   = 4;             // async pipeline depth per wave
constexpr int   NB_      = 2048;          // histogram bins
constexpr int   TILE_    = NTHREADS * 4;  // 4096 floats per streamed tile
constexpr int   NFULL_   = V_ / TILE_;    // 31 full tiles, 1024-float tail
constexpr unsigned HALF_ = (unsigned)(((unsigned long long)B_ * V_) / 2ull); // 8192000

#define INFF (__builtin_huge_valf())

// LDS byte offset of a generic pointer that points into __shared__ memory.
DEVINL unsigned lds_lo(const void* p) {
  return (unsigned)(unsigned long long)(const __attribute__((address_space(3))) char*)p;
}

DEVINL unsigned rotl32(unsigned x, int r) { return (x << r) | (x >> (32 - r)); }

// JAX gumbel noise for flat index `flat` of a (B,V) tensor under key(42).
// jax.random.key(42) -> threefry key (0, 42); counts split into two halves.
DEVINL float jax_gumbel(unsigned flat) {
  unsigned x0, x1;
  const bool lo_half = flat < HALF_;
  if (lo_half) { x0 = flat;         x1 = flat + HALF_; }
  else         { x0 = flat - HALF_; x1 = flat; }
  const unsigned k0 = 0u, k1 = 42u, k2 = k0 ^ k1 ^ 0x1BD11BDAu;
  x0 += k0; x1 += k1;
#define TF_R(a) { x0 += x1; x1 = rotl32(x1, a); x1 ^= x0; }
  TF_R(13) TF_R(15) TF_R(26) TF_R(6)  x0 += k1; x1 += k2 + 1u;
  TF_R(17) TF_R(29) TF_R(16) TF_R(24) x0 += k2; x1 += k0 + 2u;
  TF_R(13) TF_R(15) TF_R(26) TF_R(6)  x0 += k0; x1 += k1 + 3u;
  TF_R(17) TF_R(29) TF_R(16) TF_R(24) x0 += k1; x1 += k2 + 4u;
  TF_R(13) TF_R(15) TF_R(26) TF_R(6)  x0 += k2; x1 += k0 + 5u;
#undef TF_R
  const unsigned bits = lo_half ? x0 : x1;
  float u = __uint_as_float(0x3f800000u | (bits >> 9)) - 1.0f;   // [0,1)
  const float tiny = 1.1754943508222875e-38f;                    // finfo.tiny
  u = u * (1.0f - tiny) + tiny;
  u = fmaxf(u, tiny);
  return -logf(-logf(u));
}

// rep/freq/pres penalty using a sorted token list in LDS (binary search count)
DEVINL float pen_logit(const int* tk, float x, int v) {
  int lo = 0, hi = S_;
  while (lo < hi) { int m = (lo + hi) >> 1; if (tk[m] < v) lo = m + 1; else hi = m; }
  int c = 0;
  while (lo + c < S_ && tk[lo + c] == v) ++c;
  if (c) {
    x = (x > 0.f) ? (x / REP_) : (x * REP_);
    x -= FREQ_ * (float)c + PRES_;
  }
  return x;
}

DEVINL void ins3(float v, float& a, float& b, float& c) {
  if (v > a)      { c = b; b = a; a = v; }
  else if (v > b) { c = b; b = v; }
  else if (v > c) { c = v; }
}

// Stream one row (V_ floats) through LDS with a DEPTH_-deep async pipeline.
// Each wave stages its own 512B slice per tile (lane = 16B), so only the
// per-wave ASYNCcnt needs waiting -- no workgroup barriers inside. The
// functor receives 4 consecutive elements as a float4.
template <typename F>
DEVINL void stream_row(const float* __restrict__ rowbase,
                       float* __restrict__ stage, F&& f) {
  const int tid  = threadIdx.x;
  const int lane = tid & 31;
  const int w    = tid >> 5;
  float* b0 = stage + w * (DEPTH_ * 128);      // DEPTH_ x 128-float buffers
  const unsigned long long sbase = (unsigned long long)rowbase;

  auto issue = [&](int t, int sel) {
    unsigned laddr = lds_lo(b0 + sel * 128 + lane * 4);
    unsigned voff  = (unsigned)((t * TILE_ + w * 128 + lane * 4) * 4);
    asm volatile("global_load_async_to_lds_b128 %0, %1, %2"
                 :: "v"(laddr), "v"(voff), "s"(sbase) : "memory");
  };

#pragma unroll 1
  for (int p = 0; p < DEPTH_; ++p) issue(p, p);     // DEPTH_ <= NFULL_
#pragma unroll 1
  for (int t = 0; t < NFULL_; ++t) {
    const int sel = t & (DEPTH_ - 1);
    const int rem = NFULL_ - 1 - t;                 // max tiles still in flight
    if (rem >= 3)      asm volatile("s_wait_asynccnt 3" ::: "memory");
    else if (rem == 2) asm volatile("s_wait_asynccnt 2" ::: "memory");
    else if (rem == 1) asm volatile("s_wait_asynccnt 1" ::: "memory");
    else               asm volatile("s_wait_asynccnt 0" ::: "memory");
    const float4 q = *reinterpret_cast<const float4*>(b0 + sel * 128 + lane * 4);
    f(t * TILE_ + w * 128 + lane * 4, q);
    if (t + DEPTH_ < NFULL_) issue(t + DEPTH_, sel);
  }
  // tail: 1024 floats = 256 threads x float4 (126976 % 4 == 0, 16B aligned)
  if (tid < 256) {
    const int e = NFULL_ * TILE_ + tid * 4;
    const float4 q = *reinterpret_cast<const float4*>(rowbase + e);
    f(e, q);
  }
}

__global__ __launch_bounds__(NTHREADS, 1)
void Sampler_67654324846949_kernel(const float* __restrict__ logits,
                                   const int*   __restrict__ prev,
                                   float*       __restrict__ out,
                                   float*       __restrict__ plcache) {
  __shared__ int   tk[S_];                              // sorted prev tokens (8KB)
  __shared__ __align__(16) float stage[NWAVES * DEPTH_ * 128]; // staging (64KB)
  __shared__ float rA[NTHREADS], rB[NTHREADS], rC[NTHREADS], rD[NTHREADS];
  __shared__ float histF[NB_];
  __shared__ unsigned histU[NB_];
  __shared__ float keepL[NTHREADS];
  __shared__ int   keepI[NTHREADS];
  __shared__ float bc[16];
  __shared__ unsigned ncnt;

  const int row = blockIdx.x;
  const int tid = threadIdx.x;
  const float* rb = logits + (size_t)row * V_;
  const bool  use_cache = (plcache != nullptr);
  float* plrow = use_cache ? (plcache + (size_t)row * V_) : nullptr;

  // ---- load + bitonic sort prev tokens (ascending) ----
  tk[tid]        = prev[(size_t)row * S_ + tid];
  tk[tid + 1024] = prev[(size_t)row * S_ + tid + 1024];
  __syncthreads();
#pragma unroll 1
  for (int k = 2; k <= S_; k <<= 1)
#pragma unroll 1
    for (int j = k >> 1; j > 0; j >>= 1) {
      const int i = ((tid & ~(j - 1)) << 1) | (tid & (j - 1));
      const int p = i + j;
      const int a = tk[i], b = tk[p];
      const bool up = ((i & k) == 0);
      if ((a > b) == up) { tk[i] = b; tk[p] = a; }
      __syncthreads();
    }

  // ---- pass 1: penalize, track top-3 + min, optionally cache to d_ws ----
  float t1 = -INFF, t2 = -INFF, t3 = -INFF, mn = INFF;
  stream_row(rb, stage, [&](int e, float4 q) {
    float xs[4] = {q.x, q.y, q.z, q.w};
#pragma unroll
    for (int k = 0; k < 4; ++k) {
      xs[k] = pen_logit(tk, xs[k], e + k);
      ins3(xs[k], t1, t2, t3);
      mn = fminf(mn, xs[k]);
    }
    if (use_cache) {
      float4 o; o.x = xs[0]; o.y = xs[1]; o.z = xs[2]; o.w = xs[3];
      *reinterpret_cast<float4*>(plrow + e) = o;
    }
  });
  if (use_cache) __threadfence();   // make cache visible to the async DMA path
  rA[tid] = t1; rB[tid] = t2; rC[tid] = t3; rD[tid] = mn;
  __syncthreads();
  for (int s = 512; s; s >>= 1) {
    if (tid < s) {
      float a = rA[tid], b = rB[tid], c = rC[tid];
      ins3(rA[tid + s], a, b, c); ins3(rB[tid + s], a, b, c); ins3(rC[tid + s], a, b, c);
      rA[tid] = a; rB[tid] = b; rC[tid] = c;
      rD[tid] = fminf(rD[tid], rD[tid + s]);
    }
    __syncthreads();
  }
  if (tid == 0) { bc[0] = rB[0]; bc[1] = rC[0]; bc[2] = rD[0]; }
  __syncthreads();
  const float secondV = bc[0];   // avoid-top-2 threshold (remove l >= secondV)
  const float M       = bc[1];   // max of kept logits (3rd largest overall)
  const float lmin    = bc[2];

  // source + value decoder for the remaining passes
  const float* psrc = use_cache ? plrow : rb;
  auto getl = [&](float x, int v) -> float {
    return use_cache ? x : pen_logit(tk, x, v);
  };

  // ---- pass 2: S0 = sum e^a, S1 = sum a*e^a, a = (l-M)/T over kept ----
  float s0 = 0.f, s1 = 0.f;
  stream_row(psrc, stage, [&](int e, float4 q) {
    float xs[4] = {q.x, q.y, q.z, q.w};
#pragma unroll
    for (int k = 0; k < 4; ++k) {
      const float l = getl(xs[k], e + k);
      if (l < secondV) {
        const float a = (l - M) * INVT_;
        const float ex = expf(a);
        s0 += ex; s1 += a * ex;
      }
    }
  });
  rA[tid] = s0; rB[tid] = s1;
  __syncthreads();
  for (int s = 512; s; s >>= 1) {
    if (tid < s) { rA[tid] += rA[tid + s]; rB[tid] += rB[tid + s]; }
    __syncthreads();
  }
  if (tid == 0) {
    const float S0v = rA[0], S1v = rB[0];
    const float c0  = S1v / S0v;                 // dev = |c0 - a|
    const float amin = (lmin - M) * INVT_;
    float dmax = fmaxf(c0 - amin, -c0);
    dmax = fmaxf(dmax, 1e-20f) * 1.0000002f;
    bc[3] = S0v; bc[4] = c0; bc[5] = dmax;
  }
  __syncthreads();
  const float S0v   = bc[3];
  const float c0    = bc[4];
  const float dmax  = bc[5];
  const float invS0 = 1.0f / S0v;

  // ---- pass 3: coarse typicality-deviation mass histogram ----
  for (int i = tid; i < NB_; i += NTHREADS) histF[i] = 0.f;
  __syncthreads();
  stream_row(psrc, stage, [&](int e, float4 q) {
    float xs[4] = {q.x, q.y, q.z, q.w};
#pragma unroll
    for (int k = 0; k < 4; ++k) {
      const float l = getl(xs[k], e + k);
      if (l < secondV) {
        const float a = (l - M) * INVT_;
        const float ex = expf(a);
        const float dev = fabsf(c0 - a);
        int b = (int)(dev * ((float)NB_ / dmax));
        if (b > NB_ - 1) b = NB_ - 1;
        atomicAdd(&histF[b], ex * invS0);
      }
    }
  });
  __syncthreads();
  if (tid == 0) {
    float cum = 0.f; int b = -1; float carry = 0.f;
#pragma unroll 1
    for (int i = 0; i < NB_; ++i) {
      const float h = histF[i];
      if (cum + h > TYPP_) { b = i; carry = cum; break; }
      cum += h;
    }
    if (b < 0) { bc[6] = dmax * 4.f; bc[7] = dmax / NB_; bc[8] = 0.f; bc[9] = dmax * 2.f; }
    else {
      const float bw = dmax / NB_;
      bc[6] = b * bw; bc[7] = bw; bc[8] = carry; bc[9] = (b + 1) * bw;
    }
  }
  __syncthreads();
  const float tLo = bc[6], tW = bc[7], tCarry = bc[8];

  // ---- pass 3b: refine typical threshold inside the crossing bin ----
  for (int i = tid; i < NB_; i += NTHREADS) histF[i] = 0.f;
  __syncthreads();
  stream_row(psrc, stage, [&](int e, float4 q) {
    float xs[4] = {q.x, q.y, q.z, q.w};
#pragma unroll
    for (int k = 0; k < 4; ++k) {
      const float l = getl(xs[k], e + k);
      if (l < secondV) {
        const float a = (l - M) * INVT_;
        const float ex = expf(a);
        const float dev = fabsf(c0 - a);
        if (dev >= tLo && dev < tLo + tW) {
          int b = (int)((dev - tLo) * ((float)NB_ / tW));
          if (b > NB_ - 1) b = NB_ - 1;
          atomicAdd(&histF[b], ex * invS0);
        }
      }
    }
  });
  __syncthreads();
  if (tid == 0) {
    float cum = tCarry;
#pragma unroll 1
    for (int i = 0; i < NB_; ++i) {
      cum += histF[i];
      if (cum > TYPP_) { bc[9] = tLo + (i + 1) * (tW / NB_); break; }
    }
  }
  __syncthreads();
  const float Ttyp = bc[9];

  // ---- pass 4: coarse count histogram over l for top-k ----
  for (int i = tid; i < NB_; i += NTHREADS) histU[i] = 0u;
  __syncthreads();
  const float r4 = fmaxf(M - lmin, 1e-20f);
  stream_row(psrc, stage, [&](int e, float4 q) {
    float xs[4] = {q.x, q.y, q.z, q.w};
#pragma unroll
    for (int k = 0; k < 4; ++k) {
      const float l = getl(xs[k], e + k);
      if (l < secondV) {
        const float a = (l - M) * INVT_;
        if (fabsf(c0 - a) <= Ttyp) {
          int b = (int)((l - lmin) * ((float)NB_ / r4));
          if (b < 0) b = 0; if (b > NB_ - 1) b = NB_ - 1;
          atomicAdd(&histU[b], 1u);
        }
      }
    }
  });
  __syncthreads();
  if (tid == 0) {
    unsigned cum = 0; int b = -1; unsigned carry = 0;
#pragma unroll 1
    for (int i = NB_ - 1; i >= 0; --i) {
      const unsigned h = histU[i];
      if (cum + h >= (unsigned)TOPK_) { b = i; carry = cum; break; }
      cum += h;
    }
    if (b < 0) { bc[10] = 3.0e38f; bc[11] = 1.f; bc[12] = -INFF; bc[13] = 0.f; }
    else {
      const float w4 = r4 / NB_;
      bc[10] = lmin + b * w4; bc[11] = w4; bc[12] = lmin + b * w4; bc[13] = (float)carry;
    }
  }
  __syncthreads();
  const float b4lo = bc[10], w4 = bc[11], carry4 = bc[13];

  // ---- pass 4b: refine kth (50th largest) inside the crossing bin ----
  for (int i = tid; i < NB_; i += NTHREADS) histU[i] = 0u;
  __syncthreads();
  stream_row(psrc, stage, [&](int e, float4 q) {
    float xs[4] = {q.x, q.y, q.z, q.w};
#pragma unroll
    for (int k = 0; k < 4; ++k) {
      const float l = getl(xs[k], e + k);
      if (l < secondV) {
        const float a = (l - M) * INVT_;
        if (fabsf(c0 - a) <= Ttyp && l >= b4lo && l < b4lo + w4) {
          int b = (int)((l - b4lo) * ((float)NB_ / w4));
          if (b < 0) b = 0; if (b > NB_ - 1) b = NB_ - 1;
          atomicAdd(&histU[b], 1u);
        }
      }
    }
  });
  __syncthreads();
  if (tid == 0) {
    unsigned cum = (unsigned)carry4;
#pragma unroll 1
    for (int i = NB_ - 1; i >= 0; --i) {
      cum += histU[i];
      if (cum >= (unsigned)TOPK_) { bc[12] = b4lo + i * (w4 / NB_); break; }
    }
    ncnt = 0u;
  }
  __syncthreads();
  const float kth = bc[12];

  // ---- pass 5: gather survivors (typical AND >= kth) ----
  stream_row(psrc, stage, [&](int e, float4 q) {
    float xs[4] = {q.x, q.y, q.z, q.w};
#pragma unroll
    for (int k = 0; k < 4; ++k) {
      const float l = getl(xs[k], e + k);
      if (l < secondV) {
        const float a = (l - M) * INVT_;
        if (fabsf(c0 - a) <= Ttyp && l >= kth) {
          const unsigned idx = atomicAdd(&ncnt, 1u);
          if (idx < (unsigned)NTHREADS) { keepL[idx] = l; keepI[idx] = e + k; }
        }
      }
    }
  });
  __syncthreads();
  unsigned n = ncnt; if (n > (unsigned)NTHREADS) n = NTHREADS;
  if (tid >= (int)n) { keepL[tid] = -INFF; keepI[tid] = 0; }
  __syncthreads();

  // ---- bitonic sort survivors descending by logit ----
#pragma unroll 1
  for (int k = 2; k <= NTHREADS; k <<= 1)
#pragma unroll 1
    for (int j = k >> 1; j; j >>= 1) {
      if (tid < NTHREADS / 2) {
        const int i = ((tid & ~(j - 1)) << 1) | (tid & (j - 1));
        const int p = i + j;
        const bool up = ((i & k) == 0);
        const float a = keepL[i], b = keepL[p];
        if ((a < b) == up) {
          keepL[i] = b; keepL[p] = a;
          const int ti = keepI[i]; keepI[i] = keepI[p]; keepI[p] = ti;
        }
      }
      __syncthreads();
    }

  // ---- top-p + min-p + JAX gumbel-max (serial over <= ~50 survivors) ----
  if (tid == 0) {
    if (n == 0u) { out[row] = 0.f; return; }
    const float M2 = keepL[0];
    float Z = 0.f;
#pragma unroll 1
    for (unsigned i = 0; i < n; ++i) Z += expf((keepL[i] - M2) * INVT_);
    const float minl = keepL[0] + TEMP_ * logf(MINP_);  // min-p as logit bound
    float cum = 0.f, best = -INFF;
    int bi = keepI[0];
#pragma unroll 1
    for (unsigned i = 0; i < n; ++i) {
      const float L  = keepL[i];
      const float pr = expf((L - M2) * INVT_) / Z;
      if (cum <= TOPP_ && L >= minl) {           // top-p: preceding mass <= p
        const float g  = jax_gumbel((unsigned)row * (unsigned)V_ + (unsigned)keepI[i]);
        const float sc = L * INVT_ + g;
        if (sc > best) { best = sc; bi = keepI[i]; }
      }
      cum += pr;
    }
    out[row] = (float)bi;
  }
}

extern "C" void kernel_launch(void* const* d_in, const int* in_sizes, int n_in,
                              void* d_out, int out_size, void* d_ws, size_t ws_size,
                              hipStream_t stream) {
  (void)in_sizes; (void)n_in; (void)out_size;
  const float* logits = (const float*)d_in[0];
  const int*   prev   = (const int*)d_in[1];
  float*       out    = (float*)d_out;
  const size_t need   = (size_t)B_ * (size_t)V_ * sizeof(float);
  float* plcache = (ws_size >= need) ? (float*)d_ws : nullptr;
  hipLaunchKernelGGL(Sampler_67654324846949_kernel,
                     dim3(B_), dim3(NTHREADS), 0, stream, logits, prev, out, plcache);
}